// MANNCell_3315714752714
// MI455X (gfx1250) — compile-verified
//
#include <hip/hip_runtime.h>
#include <hip/hip_bf16.h>
#include <math.h>

// ---------------------------------------------------------------------------
// MANN cell for MI455X (gfx1250, wave32).
// Sizes: B=1024, N=2048, D=64, R=4, LSTM=512, IN=256, GAMMA=0.95
// ---------------------------------------------------------------------------

typedef float v2f __attribute__((ext_vector_type(2)));
typedef float v8f __attribute__((ext_vector_type(8)));

#define LDS_PITCH 68   // 64 + 4 pad -> conflict-free ds_load_b64 fragment reads

// ---------------------------------------------------------------------------
// Generic fp32 WMMA GEMM:  C[M,N] = A[M,K] @ W[N,K]^T   (no bias; bias fused
// into the following activation kernels).  A is gathered from up to 3 row-
// segments along K (handles the concat [x | r_prev | h_prev]); W from up to 2
// segments (handles [W_ih | W_hh]).  Workgroup = 4 waves, 16(M) x 64(N) tile,
// K-tile = 64 -> 16 x v_wmma_f32_16x16x4_f32 per wave per stage.
// ---------------------------------------------------------------------------
__global__ __launch_bounds__(128)
void wmma_gemm_nt(const float* __restrict__ A0, const float* __restrict__ A1,
                  const float* __restrict__ A2,
                  int ak1, int ak2, int sA0, int sA1, int sA2,
                  const float* __restrict__ W0, const float* __restrict__ W1,
                  int wk1, int sW0, int sW1,
                  float* __restrict__ C, int M, int N, int K, int ldc)
{
  __shared__ float Als[16 * LDS_PITCH];
  __shared__ float Bls[64 * LDS_PITCH];

  const int tid   = threadIdx.x;
  const int wave  = tid >> 5;
  const int lane  = tid & 31;
  const int r16   = lane & 15;
  const int hi    = lane >> 4;       // 0: lanes 0-15, 1: lanes 16-31
  const int mBase = blockIdx.y * 16;
  const int nBase = blockIdx.x * 64;

  v8f acc = {};                      // 16x16 fp32 accumulator (8 VGPRs)

  for (int kb = 0; kb < K; kb += 64) {
    // ---- stage A tile: 16 rows x 64 k (k-tiles never straddle segments) ----
    {
      const float* Aseg; int koff, stride;
      if (kb < ak1)      { Aseg = A0; koff = kb;       stride = sA0; }
      else if (kb < ak2) { Aseg = A1; koff = kb - ak1; stride = sA1; }
      else               { Aseg = A2; koff = kb - ak2; stride = sA2; }
      for (int s = tid; s < 16 * 16; s += 128) {
        int row = s >> 4, c4 = s & 15;
        float4 v = *reinterpret_cast<const float4*>(
            Aseg + (size_t)(mBase + row) * stride + koff + c4 * 4);
        *reinterpret_cast<float4*>(&Als[row * LDS_PITCH + c4 * 4]) = v;
      }
    }
    // ---- stage B tile (transposed to [n][k], k contiguous) ----
    {
      const float* Wseg; int wkoff, wstride;
      if (kb < wk1) { Wseg = W0; wkoff = kb;       wstride = sW0; }
      else          { Wseg = W1; wkoff = kb - wk1; wstride = sW1; }
      for (int s = tid; s < 64 * 16; s += 128) {
        int row = s >> 4, c4 = s & 15;
        int ng = nBase + row;
        float4 v = {0.f, 0.f, 0.f, 0.f};
        if (ng < N)
          v = *reinterpret_cast<const float4*>(
              Wseg + (size_t)ng * wstride + wkoff + c4 * 4);
        *reinterpret_cast<float4*>(&Bls[row * LDS_PITCH + c4 * 4]) = v;
      }
    }
    __syncthreads();

    // ---- 16 WMMA steps (K=4 each). ISA A-frag layout: lanes 0-15 hold
    //      K = {k0,k0+1}, lanes 16-31 hold K = {k0+2,k0+3}; B mirrors it. ----
#pragma unroll
    for (int kk = 0; kk < 16; ++kk) {
      int k0 = kk * 4 + hi * 2;
      v2f af = *reinterpret_cast<const v2f*>(&Als[r16 * LDS_PITCH + k0]);
      v2f bf = *reinterpret_cast<const v2f*>(
          &Bls[(wave * 16 + r16) * LDS_PITCH + k0]);
      acc = __builtin_amdgcn_wmma_f32_16x16x4_f32(
          /*neg_a=*/false, af, /*neg_b=*/false, bf,
          /*c_mod=*/(short)0, acc, /*reuse_a=*/false, /*reuse_b=*/false);
    }
    __syncthreads();
  }

  // ---- store: VGPR v -> M = v + 8*hi, N = lane%16 (ISA C/D layout) ----
  int n = nBase + wave * 16 + r16;
  if (n < N) {
#pragma unroll
    for (int v = 0; v < 8; ++v) {
      int m = mBase + v + 8 * hi;
      C[(size_t)m * ldc + n] = acc[v];
    }
  }
}

// ---------------------------------------------------------------------------
// LSTM gate activations (bias fused here):  c_t, h_t
// ---------------------------------------------------------------------------
__global__ __launch_bounds__(512)
void lstm_act(const float* __restrict__ gates, const float* __restrict__ b_ih,
              const float* __restrict__ b_hh, const float* __restrict__ c_prev,
              float* __restrict__ o_ht, float* __restrict__ o_ct)
{
  int b = blockIdx.x, j = threadIdx.x;
  const float* g = gates + (size_t)b * 2048;
  float ig = g[j]        + b_ih[j]        + b_hh[j];
  float fg = g[j + 512]  + b_ih[j + 512]  + b_hh[j + 512];
  float gg = g[j + 1024] + b_ih[j + 1024] + b_hh[j + 1024];
  float og = g[j + 1536] + b_ih[j + 1536] + b_hh[j + 1536];
  float si = 1.f / (1.f + __expf(-ig));
  float sf = 1.f / (1.f + __expf(-fg));
  float so = 1.f / (1.f + __expf(-og));
  float c  = sf * c_prev[(size_t)b * 512 + j] + si * tanhf(gg);
  float h  = so * tanhf(c);
  o_ct[(size_t)b * 512 + j] = c;
  o_ht[(size_t)b * 512 + j] = h;
}

// ---------------------------------------------------------------------------
// Head parameters from p (stride 528 in ws):  k, a, sigma(alpha), ||k||
// ---------------------------------------------------------------------------
__global__ __launch_bounds__(256)
void head_params(const float* __restrict__ p, const float* __restrict__ b_lin,
                 float* __restrict__ kv, float* __restrict__ av,
                 float* __restrict__ sav, float* __restrict__ knv)
{
  __shared__ float sq[256];
  int b = blockIdx.x, t = threadIdx.x;
  const float* pr = p + (size_t)b * 528;
  float kk = tanhf(pr[t] + b_lin[t]);
  kv[(size_t)b * 256 + t] = kk;
  av[(size_t)b * 256 + t] = tanhf(pr[256 + t] + b_lin[256 + t]);
  if (t < 4)
    sav[b * 4 + t] = 1.f / (1.f + __expf(-(pr[512 + t] + b_lin[512 + t])));
  sq[t] = kk * kk;
  __syncthreads();
  if (t < 4) {
    float s = 0.f;
    for (int d = 0; d < 64; ++d) s += sq[t * 64 + d];
    knv[b * 4 + t] = sqrtf(s);
  }
}

// ---------------------------------------------------------------------------
// 4-round masked argmin over wu_prev -> the R least-used slots.
// Slot 0 = global argmin (erase row).  Ties: larger index (argsort-stable).
// ---------------------------------------------------------------------------
__global__ __launch_bounds__(256)
void argmin4(const float* __restrict__ wu_prev, int* __restrict__ idx4)
{
  __shared__ float vals[2048];
  __shared__ float rv[256];
  __shared__ int   ri[256];
  int b = blockIdx.x, t = threadIdx.x;
  for (int i = 0; i < 8; ++i)
    vals[t + i * 256] = wu_prev[(size_t)b * 2048 + t + i * 256];
  __syncthreads();
  for (int round = 0; round < 4; ++round) {
    float bv = 3.4e38f; int bi = -1;
    for (int i = 0; i < 8; ++i) {
      int idx = t + i * 256;
      float v = vals[idx];
      if (v < bv || (v == bv && idx > bi)) { bv = v; bi = idx; }
    }
    rv[t] = bv; ri[t] = bi;
    __syncthreads();
    for (int s = 128; s > 0; s >>= 1) {
      if (t < s) {
        float ov = rv[t + s]; int oi = ri[t + s];
        if (ov < rv[t] || (ov == rv[t] && oi > ri[t])) { rv[t] = ov; ri[t] = oi; }
      }
      __syncthreads();
    }
    if (t == 0) { idx4[b * 4 + round] = ri[0]; vals[ri[0]] = 3.4e38f; }
    __syncthreads();
  }
}

// ---------------------------------------------------------------------------
// Pass 1 over M_prev (512 MB, read once): cosine logits + fused write-head
// update M_t (ww_t does NOT depend on the softmax).  One thread per memory row.
// ---------------------------------------------------------------------------
__global__ __launch_bounds__(256)
void mann_pass1(const float* __restrict__ M_prev, const float* __restrict__ wr_prev,
                const float* __restrict__ kv, const float* __restrict__ av,
                const float* __restrict__ sav, const float* __restrict__ knv,
                const int* __restrict__ idx4,
                float* __restrict__ o_Mt, float* __restrict__ logits,
                float* __restrict__ wwsum)
{
  __shared__ float ks[256], as_[256], sas[4], kns[4];
  __shared__ int   ids[4];
  int b = blockIdx.y, t = threadIdx.x;
  int n = blockIdx.x * 256 + t;
  ks[t]  = kv[(size_t)b * 256 + t];
  as_[t] = av[(size_t)b * 256 + t];
  if (t < 4) { sas[t] = sav[b * 4 + t]; kns[t] = knv[b * 4 + t]; ids[t] = idx4[b * 4 + t]; }
  __syncthreads();

  const float* Mrow = M_prev + ((size_t)b * 2048 + n) * 64;
  float4 m[16];
  float ip0 = 0, ip1 = 0, ip2 = 0, ip3 = 0, nrm = 0;
#pragma unroll
  for (int i = 0; i < 16; ++i) {
    m[i] = *reinterpret_cast<const float4*>(Mrow + i * 4);
    float vx[4] = {m[i].x, m[i].y, m[i].z, m[i].w};
#pragma unroll
    for (int c = 0; c < 4; ++c) {
      int d = i * 4 + c;
      float v = vx[c];
      nrm += v * v;
      ip0 += ks[d] * v;        ip1 += ks[64 + d] * v;
      ip2 += ks[128 + d] * v;  ip3 += ks[192 + d] * v;
    }
  }
  float Mn = sqrtf(nrm);
  float lg[4] = { ip0 / (kns[0] * Mn + 1e-8f), ip1 / (kns[1] * Mn + 1e-8f),
                  ip2 / (kns[2] * Mn + 1e-8f), ip3 / (kns[3] * Mn + 1e-8f) };
  float wlu = (n == ids[0] || n == ids[1] || n == ids[2] || n == ids[3]) ? 1.f : 0.f;
  float er  = (n == ids[0]) ? 0.f : 1.f;

  float ww[4], wws = 0.f;
#pragma unroll
  for (int r = 0; r < 4; ++r) {
    logits[((size_t)b * 4 + r) * 2048 + n] = lg[r];
    float wp = wr_prev[((size_t)b * 4 + r) * 2048 + n];
    ww[r] = sas[r] * wp + (1.f - sas[r]) * wlu;
    wws += ww[r];
  }
  wwsum[(size_t)b * 2048 + n] = wws;

  float* Mo = o_Mt + ((size_t)b * 2048 + n) * 64;
#pragma unroll
  for (int i = 0; i < 16; ++i) {
    int d0 = i * 4;
    float4 o;
    o.x = m[i].x * er + ww[0]*as_[d0+0]      + ww[1]*as_[64+d0+0]
                      + ww[2]*as_[128+d0+0]  + ww[3]*as_[192+d0+0];
    o.y = m[i].y * er + ww[0]*as_[d0+1]      + ww[1]*as_[64+d0+1]
                      + ww[2]*as_[128+d0+1]  + ww[3]*as_[192+d0+1];
    o.z = m[i].z * er + ww[0]*as_[d0+2]      + ww[1]*as_[64+d0+2]
                      + ww[2]*as_[128+d0+2]  + ww[3]*as_[192+d0+2];
    o.w = m[i].w * er + ww[0]*as_[d0+3]      + ww[1]*as_[64+d0+3]
                      + ww[2]*as_[128+d0+3]  + ww[3]*as_[192+d0+3];
    *reinterpret_cast<float4*>(Mo + d0) = o;
  }
}

// ---------------------------------------------------------------------------
// Softmax over N=2048, in place in the wr_t output region.  One WG per (b,r).
// ---------------------------------------------------------------------------
__global__ __launch_bounds__(256)
void softmax_rows(float* __restrict__ wr)
{
  __shared__ float red[256];
  float* base = wr + (size_t)blockIdx.x * 2048;
  int t = threadIdx.x;
  float l[8], mx = -3.4e38f;
  for (int i = 0; i < 8; ++i) { l[i] = base[t + i * 256]; mx = fmaxf(mx, l[i]); }
  red[t] = mx; __syncthreads();
  for (int s = 128; s > 0; s >>= 1) { if (t < s) red[t] = fmaxf(red[t], red[t+s]); __syncthreads(); }
  mx = red[0]; __syncthreads();
  float sum = 0.f;
  for (int i = 0; i < 8; ++i) { l[i] = __expf(l[i] - mx); sum += l[i]; }
  red[t] = sum; __syncthreads();
  for (int s = 128; s > 0; s >>= 1) { if (t < s) red[t] += red[t+s]; __syncthreads(); }
  float inv = 1.f / red[0];
  for (int i = 0; i < 8; ++i) base[t + i * 256] = l[i] * inv;
}

// ---------------------------------------------------------------------------
// Pass 2 over M_t (512 MB, read once): r_t = wr_t @ M_t (LDS-staged partial
// reduction + per-WG atomics) fused with wu_t.  Coalesced: 64 lanes sweep d.
// ---------------------------------------------------------------------------
__global__ __launch_bounds__(256)
void mann_pass2(const float* __restrict__ Mt, const float* __restrict__ wr,
                const float* __restrict__ wu_prev, const float* __restrict__ wwsum,
                float* __restrict__ o_rt, float* __restrict__ o_wu)
{
  __shared__ float wrs[4 * 256];
  __shared__ float red[4 * 256];
  int b = blockIdx.y, t = threadIdx.x;
  int nbase = blockIdx.x * 256;
#pragma unroll
  for (int i = 0; i < 4; ++i) {
    int s = t + i * 256, r = s >> 8, j = s & 255;
    wrs[r * 256 + j] = wr[((size_t)b * 4 + r) * 2048 + nbase + j];
  }
  __syncthreads();

  int g = t >> 6, d = t & 63;   // 4 n-groups x 64 d-lanes
  float a0 = 0, a1 = 0, a2 = 0, a3 = 0;
  const float* Mb = Mt + ((size_t)b * 2048 + nbase) * 64 + d;
#pragma unroll 4
  for (int jj = 0; jj < 64; ++jj) {
    int nl = g * 64 + jj;
    float mv = Mb[(size_t)nl * 64];
    a0 += wrs[nl] * mv;        a1 += wrs[256 + nl] * mv;
    a2 += wrs[512 + nl] * mv;  a3 += wrs[768 + nl] * mv;
  }
  red[t] = a0; red[256 + t] = a1; red[512 + t] = a2; red[768 + t] = a3;
  __syncthreads();
  if (t < 64) {
#pragma unroll
    for (int r = 0; r < 4; ++r) {
      float s = red[r*256 + t] + red[r*256 + 64 + t]
              + red[r*256 + 128 + t] + red[r*256 + 192 + t];
      atomicAdd(&o_rt[(size_t)b * 256 + r * 64 + t], s);
    }
  }
  int n = nbase + t;
  float wrsum = wrs[t] + wrs[256 + t] + wrs[512 + t] + wrs[768 + t];
  o_wu[(size_t)b * 2048 + n] =
      0.95f * wu_prev[(size_t)b * 2048 + n] + wrsum + wwsum[(size_t)b * 2048 + n];
}

// ---------------------------------------------------------------------------
__global__ void zero_rt(float* __restrict__ o_rt)
{
  o_rt[(size_t)blockIdx.x * 256 + threadIdx.x] = 0.f;
}

__global__ void concat_out(const float* __restrict__ o_ht,
                           const float* __restrict__ o_rt,
                           float* __restrict__ o_out)
{
  int b = blockIdx.x, t = threadIdx.x;
  for (int j = t; j < 768; j += 256)
    o_out[(size_t)b * 768 + j] =
        (j < 512) ? o_ht[(size_t)b * 512 + j] : o_rt[(size_t)b * 256 + j - 512];
}

// ---------------------------------------------------------------------------
extern "C" void kernel_launch(void* const* d_in, const int* in_sizes, int n_in,
                              void* d_out, int out_size, void* d_ws, size_t ws_size,
                              hipStream_t stream)
{
  const float* x       = (const float*)d_in[0];
  const float* M_prev  = (const float*)d_in[1];
  const float* r_prev  = (const float*)d_in[2];
  const float* h_prev  = (const float*)d_in[3];
  const float* c_prev  = (const float*)d_in[4];
  const float* wu_prev = (const float*)d_in[5];
  const float* wr_prev = (const float*)d_in[6];
  const float* W_ih    = (const float*)d_in[7];
  const float* b_ih    = (const float*)d_in[8];
  const float* W_hh    = (const float*)d_in[9];
  const float* b_hh    = (const float*)d_in[10];
  const float* W_lin   = (const float*)d_in[11];
  const float* b_lin   = (const float*)d_in[12];

  // Output layout (flat, return order): out, M_t, r_t, h_t, c_t, wu_t, wr_t
  float* oa    = (float*)d_out;
  float* o_out = oa;
  float* o_Mt  = oa + 786432;      // 1024*768
  float* o_rt  = oa + 135004160;   // + 1024*2048*64
  float* o_ht  = oa + 135266304;   // + 1024*256
  float* o_ct  = oa + 135790592;   // + 1024*512
  float* o_wu  = oa + 136314880;   // + 1024*512
  float* o_wr  = oa + 138412032;   // + 1024*2048

  // Workspace layout (~21 MB)
  float* gates = (float*)d_ws;          // 1024*2048
  float* pbuf  = gates + 2097152;       // 1024*528
  float* kv    = pbuf  + 540672;        // 1024*256
  float* av    = kv    + 262144;        // 1024*256
  float* sav   = av    + 262144;        // 1024*4
  float* knv   = sav   + 4096;          // 1024*4
  float* wwsum = knv   + 4096;          // 1024*2048
  int*   idx4  = (int*)(wwsum + 2097152); // 1024*4 ints

  // 0) zero r_t accumulation target (re-done every call: replay-safe)
  zero_rt<<<1024, 256, 0, stream>>>(o_rt);

  // 1) gates = [x | r_prev | h_prev] @ [W_ih | W_hh]^T   (M=1024,N=2048,K=1024)
  wmma_gemm_nt<<<dim3(32, 64), 128, 0, stream>>>(
      x, r_prev, h_prev, 256, 512, 256, 256, 512,
      W_ih, W_hh, 512, 512, 512,
      gates, 1024, 2048, 1024, 2048);

  // 2) LSTM activations -> h_t, c_t
  lstm_act<<<1024, 512, 0, stream>>>(gates, b_ih, b_hh, c_prev, o_ht, o_ct);

  // 3) p = h_t @ W_lin^T   (M=1024, N=516, K=512; ldc padded to 528)
  wmma_gemm_nt<<<dim3(9, 64), 128, 0, stream>>>(
      o_ht, o_ht, o_ht, 512, 512, 512, 512, 512,
      W_lin, W_lin, 512, 512, 512,
      pbuf, 1024, 516, 512, 528);

  // 4) k, a, sigma(alpha), ||k||
  head_params<<<1024, 256, 0, stream>>>(pbuf, b_lin, kv, av, sav, knv);

  // 5) 4 least-used slots + erase row
  argmin4<<<1024, 256, 0, stream>>>(wu_prev, idx4);

  // 6) pass 1 over M_prev: logits (-> wr region) + fused M_t + wwsum
  mann_pass1<<<dim3(8, 1024), 256, 0, stream>>>(
      M_prev, wr_prev, kv, av, sav, knv, idx4, o_Mt, o_wr, wwsum);

  // 7) softmax over N, in place -> wr_t
  softmax_rows<<<4096, 256, 0, stream>>>(o_wr);

  // 8) pass 2 over M_t: r_t (atomic partials) + wu_t
  mann_pass2<<<dim3(8, 1024), 256, 0, stream>>>(
      o_Mt, o_wr, wu_prev, wwsum, o_rt, o_wu);

  // 9) out = [h_t | r_t]
  concat_out<<<1024, 256, 0, stream>>>(o_ht, o_rt, o_out);
}